// EdgeClassificationGNN2_41875931136397
// MI455X (gfx1250) — compile-verified
//
#include <hip/hip_runtime.h>
#include <hip/hip_bf16.h>

#define N_NODES 100000
#define N_EDGES 800000
#define DN 128
#define DE 16
#define HDIM 128

typedef float v2f __attribute__((ext_vector_type(2)));
typedef float v8f __attribute__((ext_vector_type(8)));

// =========================================================================
// f32 WMMA 16x16x4 fragment conventions:
// A (16x4): lanes 0-15 -> K = k0,k0+1 ; lanes 16-31 -> K = k0+2,k0+3
// B (4x16): mirrored; C/D: VGPR v -> row v (+8 for hi lanes), col lane&15
// =========================================================================

// ---- single M-tile: OUT[16 x NCOLS] = act(As[16 x K] @ W + bias) --------
template <int ACT, int K, int LDA, int LDW, int LDOUT, int NCOLS, bool HAS_BIAS>
__device__ __forceinline__ void wave_gemm16(const float* __restrict__ As,
                                            const float* __restrict__ W,
                                            const float* __restrict__ bias,
                                            float* __restrict__ Out, int lane) {
  const int n  = lane & 15;
  const int hi = lane >> 4;
  const float* Ap = As + n * LDA + 2 * hi;
  const float* Wp = W + 2 * hi * LDW + n;
#pragma unroll
  for (int n0 = 0; n0 < NCOLS; n0 += 16) {
    v8f c = {};
#pragma unroll 8
    for (int k0 = 0; k0 < K; k0 += 4) {
      v2f a;
      a.x = Ap[k0];
      a.y = Ap[k0 + 1];
      v2f b;
      b.x = Wp[k0 * LDW + n0];
      b.y = Wp[(k0 + 1) * LDW + n0];
      c = __builtin_amdgcn_wmma_f32_16x16x4_f32(false, a, false, b, (short)0, c, false, false);
    }
    float bv = HAS_BIAS ? bias[n0 + n] : 0.0f;
    float* Op = Out + n0 + n;
#pragma unroll
    for (int v = 0; v < 8; ++v) {
      float x = c[v] + bv;
      if (ACT == 1) x = fmaxf(x, 0.0f);
      if (ACT == 2) x = tanhf(x);
      Op[(v + 8 * hi) * LDOUT] = x;
    }
  }
}

// ---- dual M-tile: OUT[32 x NCOLS] = act(As[32 x K] @ W + bias) ----------
// Each B fragment feeds two WMMAs -> half the global B traffic per FLOP.
template <int ACT, int K, int LDA, int LDW, int LDOUT, int NCOLS, bool HAS_BIAS>
__device__ __forceinline__ void wave_gemm32(const float* __restrict__ As,
                                            const float* __restrict__ W,
                                            const float* __restrict__ bias,
                                            float* __restrict__ Out, int lane) {
  const int n  = lane & 15;
  const int hi = lane >> 4;
  const float* Ap0 = As + n * LDA + 2 * hi;
  const float* Ap1 = Ap0 + 16 * LDA;
  const float* Wp  = W + 2 * hi * LDW + n;
#pragma unroll
  for (int n0 = 0; n0 < NCOLS; n0 += 16) {
    v8f c0 = {};
    v8f c1 = {};
#pragma unroll 4
    for (int k0 = 0; k0 < K; k0 += 4) {
      v2f b;
      b.x = Wp[k0 * LDW + n0];
      b.y = Wp[(k0 + 1) * LDW + n0];
      v2f a0;
      a0.x = Ap0[k0];
      a0.y = Ap0[k0 + 1];
      v2f a1;
      a1.x = Ap1[k0];
      a1.y = Ap1[k0 + 1];
      c0 = __builtin_amdgcn_wmma_f32_16x16x4_f32(false, a0, false, b, (short)0, c0, false, false);
      c1 = __builtin_amdgcn_wmma_f32_16x16x4_f32(false, a1, false, b, (short)0, c1, false, false);
    }
    float bv = HAS_BIAS ? bias[n0 + n] : 0.0f;
    float* Op0 = Out + n0 + n;
    float* Op1 = Op0 + 16 * LDOUT;
#pragma unroll
    for (int v = 0; v < 8; ++v) {
      int ro = (v + 8 * hi) * LDOUT;
      float x0 = c0[v] + bv;
      float x1 = c1[v] + bv;
      if (ACT == 1) { x0 = fmaxf(x0, 0.0f); x1 = fmaxf(x1, 0.0f); }
      if (ACT == 2) { x0 = tanhf(x0);       x1 = tanhf(x1); }
      Op0[ro] = x0;
      Op1[ro] = x1;
    }
  }
}

// ---- dense node GEMM: Out[r,:] = X[r,:] @ W  (128x128) -------------------
#define SA_LD 132  // 132 % 64 == 4 -> conflict-free A-frag reads, 16B-aligned rows
__global__ void __launch_bounds__(128) node_gemm(const float* __restrict__ X,
                                                 const float* __restrict__ W,
                                                 float* __restrict__ Out, int nrows) {
  __shared__ float sA[4][16 * SA_LD];
  int wave = threadIdx.x >> 5;
  int lane = threadIdx.x & 31;
  int row0 = (blockIdx.x * 4 + wave) << 4;
  if (row0 >= nrows) return;  // tile-uniform exit: wave-local LDS only, no barriers
  float* As = sA[wave];
  const float* Xr = X + (size_t)row0 * DN;
#pragma unroll
  for (int j = 0; j < 16; ++j) {  // stage 16x128 tile, coalesced float4
    int idx = lane * 16 + j;
    int r = idx >> 5, c4 = idx & 31;
    *(float4*)&As[r * SA_LD + c4 * 4] = ((const float4*)(Xr + r * DN))[c4];
  }
  wave_gemm16<0, DN, SA_LD, HDIM, HDIM, HDIM, false>(As, W, nullptr,
                                                     Out + (size_t)row0 * HDIM, lane);
}

// ---- GCN normalization + aggregation ------------------------------------
__global__ void zero_f32(float* p, int n) {
  int i = blockIdx.x * blockDim.x + threadIdx.x;
  if (i < n) p[i] = 0.0f;
}
__global__ void deg_count(const int* __restrict__ dst, float* __restrict__ deg, int e) {
  int i = blockIdx.x * blockDim.x + threadIdx.x;
  if (i < e) atomicAdd(&deg[dst[i]], 1.0f);
}
__global__ void make_dinv(const float* __restrict__ deg, float* __restrict__ dinv, int n) {
  int i = blockIdx.x * blockDim.x + threadIdx.x;
  if (i < n) dinv[i] = rsqrtf(deg[i] + 1.0f);  // +1 = self loop; always > 0
}
// out = xw * dinv^2 (self-loop term) + bias   (float4 per thread)
__global__ void agg_init(const float* __restrict__ xw, const float* __restrict__ dinv,
                         const float* __restrict__ bias, float* __restrict__ out, int n) {
  int i = blockIdx.x * blockDim.x + threadIdx.x;
  if (i >= n) return;
  int r = i >> 5, c4 = (i & 31) << 2;
  float di = dinv[r];
  float s = di * di;
  float4 v = *(const float4*)(xw + (size_t)r * HDIM + c4);
  float4 o;
  o.x = v.x * s + bias[c4 + 0];
  o.y = v.y * s + bias[c4 + 1];
  o.z = v.z * s + bias[c4 + 2];
  o.w = v.w * s + bias[c4 + 3];
  *(float4*)(out + (size_t)r * HDIM + c4) = o;
}
// out[dst] += xw[src] * dinv[src]*dinv[dst]   (L2-resident atomics)
__global__ void agg_edges(const float* __restrict__ xw, const float* __restrict__ dinv,
                          const int* __restrict__ src, const int* __restrict__ dst,
                          float* __restrict__ out, int n) {
  int i = blockIdx.x * blockDim.x + threadIdx.x;
  if (i >= n) return;
  int e = i >> 5, c4 = (i & 31) << 2;
  int s = src[e], d = dst[e];
  float norm = dinv[s] * dinv[d];
  float4 v = *(const float4*)(xw + (size_t)s * HDIM + c4);
  float* o = out + (size_t)d * HDIM + c4;
  atomicAdd(o + 0, v.x * norm);
  atomicAdd(o + 1, v.y * norm);
  atomicAdd(o + 2, v.z * norm);
  atomicAdd(o + 3, v.w * norm);
}

// ---- fused edge pipeline: encoder + concat + 3-layer MLP -----------------
// 32 edges per wave (dual M-tiles); dynamic LDS (66.5 KB > legacy 64 KB cap,
// fine on CDNA5's 320 KB WGP LDS).
#define BX_LD 388  // >=384, %64==4 -> conflict-free, rows 16B-aligned
#define BY_LD 132
#define ET 32      // edges per wave
#define EDGE_SMEM ((ET * BX_LD + ET * BY_LD) * 4)
__global__ void __launch_bounds__(32) edge_mlp(
    const float* __restrict__ h, const int* __restrict__ src, const int* __restrict__ dst,
    const float* __restrict__ eattr,
    const float* __restrict__ We1, const float* __restrict__ be1,
    const float* __restrict__ We2, const float* __restrict__ be2,
    const float* __restrict__ Wc1, const float* __restrict__ bc1,
    const float* __restrict__ Wc2, const float* __restrict__ bc2,
    const float* __restrict__ Wc3, const float* __restrict__ bc3,
    float* __restrict__ out) {
  extern __shared__ float smem[];
  float* bufX = smem;                 // [ET x BX_LD] concat: [h_src | h_dst | e]
  float* bufY = smem + ET * BX_LD;    // [ET x BY_LD]
  int lane = threadIdx.x;
  int e0 = blockIdx.x * ET;  // E % 32 == 0 -> EXEC all ones everywhere

  // edge_attr tile [32 x 16] -> bufX cols 0..15 (scratch use before gathers)
#pragma unroll
  for (int j = 0; j < 4; ++j) {
    int idx = lane * 4 + j;          // 128 float4
    int r = idx >> 2, c4 = idx & 3;
    *(float4*)&bufX[r * BX_LD + c4 * 4] =
        ((const float4*)eattr)[(size_t)(e0 + r) * 4 + c4];
  }
  __syncthreads();
  // e1 = relu(EA @ We1 + be1)            [32 x 128] -> bufY
  wave_gemm32<1, DE, BX_LD, HDIM, BY_LD, HDIM, true>(bufX, We1, be1, bufY, lane);
  __syncthreads();
  // e  = e1 @ We2 + be2  -> concat slice K=256..383
  wave_gemm32<0, HDIM, BY_LD, HDIM, BX_LD, HDIM, true>(bufY, We2, be2, bufX + 256, lane);
  // gather h[src] cols 0..63 -> K=0..63, h[dst] cols 0..63 -> K=128..191 (L2-resident, float4)
  {
    int r = lane;                    // one edge per lane
    int s = src[e0 + r], d = dst[e0 + r];
    const float4* hs = (const float4*)(h + (size_t)s * HDIM);
    const float4* hd = (const float4*)(h + (size_t)d * HDIM);
    float4* ox = (float4*)&bufX[r * BX_LD];
#pragma unroll
    for (int j = 0; j < 16; ++j) {
      ox[j] = hs[j];            // cols   0..63
      ox[32 + j] = hd[j];       // cols 128..191
    }
  }
  __syncthreads();
  // z1 = tanh(edge_repr @ Wc1 + bc1)     K=384 -> bufY
  wave_gemm32<2, 3 * HDIM, BX_LD, HDIM, BY_LD, HDIM, true>(bufX, Wc1, bc1, bufY, lane);
  __syncthreads();
  // z2 = tanh(z1 @ Wc2 + bc2)            [32 x 64] -> bufX cols 0..63
  wave_gemm32<2, HDIM, BY_LD, 64, BX_LD, 64, true>(bufY, Wc2, bc2, bufX, lane);
  __syncthreads();
  // logits = z2 @ Wc3 + bc3              one edge per lane (all 32 lanes)
  {
    float acc = bc3[0];
#pragma unroll 8
    for (int k = 0; k < 64; ++k) acc += bufX[lane * BX_LD + k] * Wc3[k];
    out[e0 + lane] = acc;
  }
}

extern "C" void kernel_launch(void* const* d_in, const int* in_sizes, int n_in,
                              void* d_out, int out_size, void* d_ws, size_t ws_size,
                              hipStream_t stream) {
  (void)in_sizes; (void)n_in; (void)out_size; (void)ws_size;
  const float* x     = (const float*)d_in[0];
  const int*   ei    = (const int*)d_in[1];
  const int*   src   = ei;
  const int*   dst   = ei + N_EDGES;
  const float* eattr = (const float*)d_in[2];
  const float* W1  = (const float*)d_in[3];
  const float* b1  = (const float*)d_in[4];
  const float* W2  = (const float*)d_in[5];
  const float* b2  = (const float*)d_in[6];
  const float* We1 = (const float*)d_in[7];
  const float* be1 = (const float*)d_in[8];
  const float* We2 = (const float*)d_in[9];
  const float* be2 = (const float*)d_in[10];
  const float* Wc1 = (const float*)d_in[11];
  const float* bc1 = (const float*)d_in[12];
  const float* Wc2 = (const float*)d_in[13];
  const float* bc2 = (const float*)d_in[14];
  const float* Wc3 = (const float*)d_in[15];
  const float* bc3 = (const float*)d_in[16];
  float* out = (float*)d_out;

  float* ws   = (float*)d_ws;
  float* deg  = ws;                                // N
  float* dinv = deg + N_NODES;                     // N
  float* bufA = dinv + N_NODES;                    // N*H
  float* bufB = bufA + (size_t)N_NODES * HDIM;     // N*H

  const int TPB = 256;
  int gN  = (N_NODES + TPB - 1) / TPB;
  int gE  = (N_EDGES + TPB - 1) / TPB;
  int gNH = (N_NODES * 32 + TPB - 1) / TPB;
  int gEH = (N_EDGES * 32 + TPB - 1) / TPB;
  int tiles = (N_NODES + 15) / 16;
  int gGemm = (tiles + 3) / 4;

  // degrees / symmetric norm
  zero_f32 <<<gN, TPB, 0, stream>>>(deg, N_NODES);
  deg_count<<<gE, TPB, 0, stream>>>(dst, deg, N_EDGES);
  make_dinv<<<gN, TPB, 0, stream>>>(deg, dinv, N_NODES);

  // conv1: h1 = agg(x@W1) + b1
  node_gemm<<<gGemm, 128, 0, stream>>>(x, W1, bufA, N_NODES);
  agg_init <<<gNH, TPB, 0, stream>>>(bufA, dinv, b1, bufB, N_NODES * 32);
  agg_edges<<<gEH, TPB, 0, stream>>>(bufA, dinv, src, dst, bufB, N_EDGES * 32);

  // conv2: h2 = agg(h1@W2) + b2
  node_gemm<<<gGemm, 128, 0, stream>>>(bufB, W2, bufA, N_NODES);
  agg_init <<<gNH, TPB, 0, stream>>>(bufA, dinv, b2, bufB, N_NODES * 32);
  agg_edges<<<gEH, TPB, 0, stream>>>(bufA, dinv, src, dst, bufB, N_EDGES * 32);

  // fused edge encoder + classifier (32 edges / wave)
  edge_mlp<<<N_EDGES / ET, 32, EDGE_SMEM, stream>>>(bufB, src, dst, eattr,
                                                    We1, be1, We2, be2, Wc1, bc1,
                                                    Wc2, bc2, Wc3, bc3, out);
}